// SAGE_7146825581283
// MI455X (gfx1250) — compile-verified
//
#include <hip/hip_runtime.h>

typedef __attribute__((ext_vector_type(2))) float v2f;
typedef __attribute__((ext_vector_type(4))) float v4f;
typedef __attribute__((ext_vector_type(8))) float v8f;

#define N_NODES 100000
#define N_EDGES 1600000
#define IN_F 128
#define HID 128
#define NCLS 64

#define SCAN_BLOCK 256
#define SCAN_NBLOCKS ((N_NODES + SCAN_BLOCK - 1) / SCAN_BLOCK)  // 391

// ---------------- zero fill (ints) ----------------
__global__ void zero_i_kernel(int* __restrict__ p, int n) {
    int i = blockIdx.x * blockDim.x + threadIdx.x;
    if (i < n) p[i] = 0;
}

// ---------------- degree histogram (int) ----------------
__global__ void degree_kernel(const int* __restrict__ dst, int* __restrict__ deg,
                              int n_edges) {
    int e = blockIdx.x * blockDim.x + threadIdx.x;
    if (e < n_edges) atomicAdd(&deg[dst[e]], 1);
}

__global__ void invdeg_kernel(const int* __restrict__ deg, float* __restrict__ inv_deg,
                              int n) {
    int i = blockIdx.x * blockDim.x + threadIdx.x;
    if (i < n) inv_deg[i] = 1.0f / fmaxf((float)deg[i], 1.0f);
}

// ---------------- exclusive scan (3-phase, N = 100000) ----------------
__global__ void scan_block_kernel(const int* __restrict__ in, int* __restrict__ out,
                                  int* __restrict__ blockSums, int n) {
    __shared__ int tmp[SCAN_BLOCK];
    int i = blockIdx.x * SCAN_BLOCK + threadIdx.x;
    int v = (i < n) ? in[i] : 0;
    tmp[threadIdx.x] = v;
    __syncthreads();
#pragma unroll
    for (int off = 1; off < SCAN_BLOCK; off <<= 1) {
        int t = (threadIdx.x >= off) ? tmp[threadIdx.x - off] : 0;
        __syncthreads();
        tmp[threadIdx.x] += t;
        __syncthreads();
    }
    if (i < n) out[i] = tmp[threadIdx.x] - v;                 // exclusive
    if (threadIdx.x == SCAN_BLOCK - 1) blockSums[blockIdx.x] = tmp[threadIdx.x];
}

__global__ void scan_sums_kernel(int* __restrict__ sums, int nb) {
    __shared__ int tmp[512];
    int v = (threadIdx.x < nb) ? sums[threadIdx.x] : 0;
    tmp[threadIdx.x] = v;
    __syncthreads();
#pragma unroll
    for (int off = 1; off < 512; off <<= 1) {
        int t = (threadIdx.x >= off) ? tmp[threadIdx.x - off] : 0;
        __syncthreads();
        tmp[threadIdx.x] += t;
        __syncthreads();
    }
    if (threadIdx.x < nb) sums[threadIdx.x] = tmp[threadIdx.x] - v;  // exclusive
}

__global__ void scan_add_kernel(int* __restrict__ out, const int* __restrict__ blockSums,
                                int n) {
    int i = blockIdx.x * SCAN_BLOCK + threadIdx.x;
    if (i < n) out[i] += blockSums[blockIdx.x];
}

__global__ void set_tail_kernel(int* __restrict__ row_start) {
    row_start[N_NODES] = N_EDGES;
}

// ---------------- CSR fill: edge_src[row_start[dst]+slot] = src ----------------
__global__ void fill_kernel(const int* __restrict__ src, const int* __restrict__ dst,
                            const int* __restrict__ row_start, int* __restrict__ counter,
                            int* __restrict__ edge_src, int n_edges) {
    int e = blockIdx.x * blockDim.x + threadIdx.x;
    if (e >= n_edges) return;
    int d = dst[e];
    int pos = atomicAdd(&counter[d], 1);
    edge_src[row_start[d] + pos] = src[e];
}

// ---------------- mean aggregation: wave-per-node gather ----------------
// 32 lanes x float4 = 128 features per row. Rows of h (51.2 MB) are
// L2-resident on MI455X (192 MB L2): each edge costs one coalesced 512 B
// L2 read instead of 128 f32 atomics. inv_deg is folded in here.
__global__ void aggregate_gather_kernel(const float* __restrict__ h,
                                        const int* __restrict__ row_start,
                                        const int* __restrict__ edge_src,
                                        const float* __restrict__ inv_deg,
                                        float* __restrict__ agg) {
    int gid  = blockIdx.x * blockDim.x + threadIdx.x;
    int node = gid >> 5;
    if (node >= N_NODES) return;
    int c = (gid & 31) << 2;

    int e   = row_start[node];
    int end = row_start[node + 1];
    v4f acc = (v4f){0.f, 0.f, 0.f, 0.f};
    for (; e + 1 < end; e += 2) {           // 2-way unroll for load overlap
        int s0 = edge_src[e];
        int s1 = edge_src[e + 1];
        v4f v0 = *(const v4f*)(h + s0 * 128 + c);
        v4f v1 = *(const v4f*)(h + s1 * 128 + c);
        acc += v0;
        acc += v1;
    }
    if (e < end) {
        int s0 = edge_src[e];
        acc += *(const v4f*)(h + s0 * 128 + c);
    }
    float sc = inv_deg[node];
    acc *= sc;
    *(v4f*)(agg + node * 128 + c) = acc;
}

// ---------------- fused SAGE GEMM via V_WMMA_F32_16X16X4_F32 ----------------
// out[r, c] = relu?( Hs[r,:]@Wself[:,c] + Agg[r,:]@Wneigh[:,c] + b[c] )
// One wave32 computes one 16x16 output tile; K = 128 fully unrolled.
// A frag (16x4 f32): lane L holds M = L&15; upper half-wave carries K+2/K+3.
// B frag (4x16 f32): lane L holds N = L&15; same K striping.
// C/D (16x16 f32, 8 VGPRs): VGPR j -> row j (lanes 0-15) / row j+8 (lanes 16-31).
template <int M_DIM, bool RELU>
__global__ void sage_gemm_kernel(const float* __restrict__ Hs,
                                 const float* __restrict__ Agg,
                                 const float* __restrict__ Wself,
                                 const float* __restrict__ Wneigh,
                                 const float* __restrict__ bias,
                                 float* __restrict__ out) {
    constexpr int K = 128;
    constexpr int colTiles = M_DIM / 16;

    const int lane = threadIdx.x & 31;
    const int wave = (blockIdx.x * blockDim.x + threadIdx.x) >> 5;
    const int rowTile = wave / colTiles;
    const int colTile = wave % colTiles;
    if (rowTile * 16 >= N_NODES) return;  // wave-uniform; EXEC stays all-ones

    const int mn  = lane & 15;            // M index for A, N index for B
    const int kh  = (lane >> 4) << 1;     // 0 for lanes 0-15, 2 for lanes 16-31
    const int row = rowTile * 16 + mn;

    const float* __restrict__ hrow = Hs  + row * K;
    const float* __restrict__ arow = Agg + row * K;
    const float* __restrict__ ws   = Wself  + colTile * 16 + mn;  // stride M_DIM
    const float* __restrict__ wn   = Wneigh + colTile * 16 + mn;

    v8f c = {};

#pragma unroll
    for (int k = 0; k < K; k += 4) {
        v2f a, b;
        a.x = hrow[k + kh];
        a.y = hrow[k + kh + 1];
        b.x = ws[(k + kh) * M_DIM];
        b.y = ws[(k + kh + 1) * M_DIM];
        c = __builtin_amdgcn_wmma_f32_16x16x4_f32(false, a, false, b,
                                                  (short)0, c, false, false);
    }
#pragma unroll
    for (int k = 0; k < K; k += 4) {
        v2f a, b;
        a.x = arow[k + kh];
        a.y = arow[k + kh + 1];
        b.x = wn[(k + kh) * M_DIM];
        b.y = wn[(k + kh + 1) * M_DIM];
        c = __builtin_amdgcn_wmma_f32_16x16x4_f32(false, a, false, b,
                                                  (short)0, c, false, false);
    }

    const float bv    = bias[colTile * 16 + mn];
    const int   rbase = rowTile * 16 + ((lane >> 4) << 3);  // +0 or +8
#pragma unroll
    for (int j = 0; j < 8; ++j) {
        float v = c[j] + bv;
        if (RELU) v = fmaxf(v, 0.0f);
        out[(rbase + j) * M_DIM + (colTile * 16 + mn)] = v;
    }
}

// ---------------- orchestration ----------------
extern "C" void kernel_launch(void* const* d_in, const int* in_sizes, int n_in,
                              void* d_out, int out_size, void* d_ws, size_t ws_size,
                              hipStream_t stream) {
    (void)in_sizes; (void)n_in; (void)out_size; (void)ws_size;

    const float* features = (const float*)d_in[0];
    const float* Wself0   = (const float*)d_in[1];
    const float* Wneigh0  = (const float*)d_in[2];
    const float* b0       = (const float*)d_in[3];
    const float* Wself1   = (const float*)d_in[4];
    const float* Wneigh1  = (const float*)d_in[5];
    const float* b1       = (const float*)d_in[6];
    const float* Wself2   = (const float*)d_in[7];
    const float* Wneigh2  = (const float*)d_in[8];
    const float* b2       = (const float*)d_in[9];
    const int*   src      = (const int*)d_in[10];
    const int*   dst      = (const int*)d_in[11];
    float*       out      = (float*)d_out;

    char* ws = (char*)d_ws;
    size_t off = 0;
    int*   deg_i     = (int*)(ws + off);  off += (size_t)N_NODES * 4;        // 400000
    int*   counter   = (int*)(ws + off);  off += (size_t)N_NODES * 4;        // 400000
    int*   row_start = (int*)(ws + off);  off += ((size_t)N_NODES + 4) * 4;  // 400016
    int*   blockSums = (int*)(ws + off);  off += 512 * 4;                    // 2048
    float* inv_deg   = (float*)(ws + off); off += (size_t)N_NODES * 4;       // 400000
    int*   edge_src  = (int*)(ws + off);  off += (size_t)N_EDGES * 4;        // 6.4 MB
    float* agg       = (float*)(ws + off); off += (size_t)N_NODES * 128 * 4; // 51.2 MB
    float* h1        = (float*)(ws + off); off += (size_t)N_NODES * 128 * 4; // 51.2 MB
    float* h2        = (float*)(ws + off); off += (size_t)N_NODES * 128 * 4; // 51.2 MB

    const int nodeBlocks = (N_NODES + 255) / 256;
    const int edgeBlocks = (N_EDGES + 255) / 256;

    // ---- CSR build (once, reused by all 3 layers) ----
    zero_i_kernel<<<nodeBlocks, 256, 0, stream>>>(deg_i, N_NODES);
    degree_kernel<<<edgeBlocks, 256, 0, stream>>>(dst, deg_i, N_EDGES);
    invdeg_kernel<<<nodeBlocks, 256, 0, stream>>>(deg_i, inv_deg, N_NODES);

    scan_block_kernel<<<SCAN_NBLOCKS, SCAN_BLOCK, 0, stream>>>(
        deg_i, row_start, blockSums, N_NODES);
    scan_sums_kernel<<<1, 512, 0, stream>>>(blockSums, SCAN_NBLOCKS);
    scan_add_kernel<<<SCAN_NBLOCKS, SCAN_BLOCK, 0, stream>>>(
        row_start, blockSums, N_NODES);
    set_tail_kernel<<<1, 1, 0, stream>>>(row_start);

    zero_i_kernel<<<nodeBlocks, 256, 0, stream>>>(counter, N_NODES);
    fill_kernel<<<edgeBlocks, 256, 0, stream>>>(src, dst, row_start, counter,
                                                edge_src, N_EDGES);

    const int aggBlocks = (N_NODES * 32 + 255) / 256;  // wave per node, 8 waves/block

    // ---- layer 0: features -> h1 (ReLU) ----
    aggregate_gather_kernel<<<aggBlocks, 256, 0, stream>>>(
        features, row_start, edge_src, inv_deg, agg);
    {
        int blocks = (N_NODES / 16) * (HID / 16) / 8;  // 50000 waves / 8
        sage_gemm_kernel<HID, true><<<blocks, 256, 0, stream>>>(
            features, agg, Wself0, Wneigh0, b0, h1);
    }

    // ---- layer 1: h1 -> h2 (ReLU) ----
    aggregate_gather_kernel<<<aggBlocks, 256, 0, stream>>>(
        h1, row_start, edge_src, inv_deg, agg);
    {
        int blocks = (N_NODES / 16) * (HID / 16) / 8;
        sage_gemm_kernel<HID, true><<<blocks, 256, 0, stream>>>(
            h1, agg, Wself1, Wneigh1, b1, h2);
    }

    // ---- layer 2: h2 -> out (no ReLU) ----
    aggregate_gather_kernel<<<aggBlocks, 256, 0, stream>>>(
        h2, row_start, edge_src, inv_deg, agg);
    {
        int blocks = (N_NODES / 16) * (NCLS / 16) / 8;  // 25000 waves / 8
        sage_gemm_kernel<NCLS, false><<<blocks, 256, 0, stream>>>(
            h2, agg, Wself2, Wneigh2, b2, out);
    }
}